// ASGRA_19250043421160
// MI455X (gfx1250) — compile-verified
//
#include <hip/hip_runtime.h>
#include <hip/hip_bf16.h>

#define HID 256
#define HEADS 4
#define CHEAD 64
#define IN0 288
#define NREL 51
#define GCNT 64
#define NCLS 8
#define NEG_SLOPE 0.2f

typedef __attribute__((ext_vector_type(2))) float v2f;
typedef __attribute__((ext_vector_type(8))) float v8f;

// Order-preserving float<->uint map so we can do float segment-max with atomicMax(u32).
__device__ __forceinline__ unsigned fmap(float f) {
  unsigned u = __float_as_uint(f);
  return (u & 0x80000000u) ? ~u : (u | 0x80000000u);
}
__device__ __forceinline__ float funmap(unsigned u) {
  return (u & 0x80000000u) ? __uint_as_float(u ^ 0x80000000u) : __uint_as_float(~u);
}

__global__ void fill_u32_kernel(unsigned* __restrict__ p, long long n, unsigned v) {
  long long i = (long long)blockIdx.x * blockDim.x + threadIdx.x;
  if (i < n) p[i] = v;
}

// h0[i, 0:256] = tok_emb[tok(i)] ; h0[i, 256:288] = bbox(i) @ bpW + bpb
__global__ void embed_kernel(const float* __restrict__ x, const float* __restrict__ tok_emb,
                             const float* __restrict__ bpW, const float* __restrict__ bpb,
                             float* __restrict__ h0, int N_) {
  long long idx = (long long)blockIdx.x * blockDim.x + threadIdx.x;
  if (idx >= (long long)N_ * IN0) return;
  int i = (int)(idx / IN0);
  int f = (int)(idx % IN0);
  if (f < 256) {
    int tok = (int)x[(size_t)i * 5];
    h0[idx] = tok_emb[(size_t)tok * 256 + f];
  } else {
    int j = f - 256;
    float acc = bpb[j];
#pragma unroll
    for (int k = 0; k < 4; ++k) acc += x[(size_t)i * 5 + 1 + k] * bpW[k * 32 + j];
    h0[idx] = acc;
  }
}

// efeat[r, f] = sum_k rel_emb[r,k] * We[k,f]   (51 x 64 @ 64 x 256)
__global__ __launch_bounds__(256) void efeat_kernel(const float* __restrict__ rel_emb,
                                                    const float* __restrict__ We,
                                                    float* __restrict__ efeat) {
  int r = blockIdx.x, f = threadIdx.x;
  __shared__ float rr[CHEAD];
  if (f < CHEAD) rr[f] = rel_emb[(size_t)r * CHEAD + f];
  __syncthreads();
  float acc = 0.f;
#pragma unroll 8
  for (int k = 0; k < CHEAD; ++k) acc += rr[k] * We[(size_t)k * HID + f];
  efeat[(size_t)r * HID + f] = acc;
}

// fp32 WMMA GEMM: Out(MxHID) = A(MxK) @ W(KxHID) + bias.  One wave -> 16x64 strip,
// K-loop step 4 with V_WMMA_F32_16X16X4_F32, A fragment reused across 4 B tiles.
__global__ __launch_bounds__(256) void gemm_wmma_kernel(const float* __restrict__ A,
                                                        const float* __restrict__ W,
                                                        const float* __restrict__ bias,
                                                        float* __restrict__ Out,
                                                        int M, int K) {
  int wave = blockIdx.x * 8 + (threadIdx.x >> 5);
  int lane = threadIdx.x & 31;
  int tm = (wave >> 2) * 16;   // 4 column strips of 64 per row tile
  int tn = (wave & 3) * 64;
  if (tm + 16 > M) return;
  int lrow = lane & 15;
  int koff = (lane >> 4) << 1;  // lanes 0-15: K pair {0,1}; lanes 16-31: {2,3}
  const float* Arow = A + (size_t)(tm + lrow) * K;
  v8f acc0 = {}, acc1 = {}, acc2 = {}, acc3 = {};
  for (int k = 0; k < K; k += 4) {
    v2f a;
    a.x = Arow[k + koff];
    a.y = Arow[k + koff + 1];
    const float* B0 = W + (size_t)(k + koff) * HID + tn + lrow;
    const float* B1 = B0 + HID;
    v2f b;
    b.x = B0[0];  b.y = B1[0];
    acc0 = __builtin_amdgcn_wmma_f32_16x16x4_f32(false, a, false, b, (short)0, acc0, false, false);
    b.x = B0[16]; b.y = B1[16];
    acc1 = __builtin_amdgcn_wmma_f32_16x16x4_f32(false, a, false, b, (short)0, acc1, false, false);
    b.x = B0[32]; b.y = B1[32];
    acc2 = __builtin_amdgcn_wmma_f32_16x16x4_f32(false, a, false, b, (short)0, acc2, false, false);
    b.x = B0[48]; b.y = B1[48];
    acc3 = __builtin_amdgcn_wmma_f32_16x16x4_f32(false, a, false, b, (short)0, acc3, false, false);
  }
  int grow = tm + ((lane >> 4) << 3);  // C/D: VGPR v holds M = v (+8 for upper half-wave)
  int gcol = tn + lrow;
#pragma unroll
  for (int j = 0; j < 4; ++j) {
    v8f acc = (j == 0) ? acc0 : (j == 1) ? acc1 : (j == 2) ? acc2 : acc3;
    float bv = bias[gcol + j * 16];
#pragma unroll
    for (int v = 0; v < 8; ++v)
      Out[(size_t)(grow + v) * HID + gcol + j * 16] = acc[v] + bv;
  }
}

// Wave per edge: m = xl[src]+xr[dst]+efeat[attr]; score[h] = sum_c lrelu(m)*att ;
// also atomicMax of ordered-mapped score into per-(dst,head) segment max.
__global__ __launch_bounds__(256) void edge_score_kernel(const int* __restrict__ ei,
                                                         const int* __restrict__ ea,
                                                         const float* __restrict__ xl,
                                                         const float* __restrict__ xr,
                                                         const float* __restrict__ efeat,
                                                         const float* __restrict__ att,
                                                         float* __restrict__ score,
                                                         unsigned* __restrict__ segmax,
                                                         int E_) {
  int wave = blockIdx.x * 8 + (threadIdx.x >> 5);
  int lane = threadIdx.x & 31;
  if (wave >= E_) return;
  int s = ei[wave];
  int d = ei[(size_t)E_ + wave];
  int r = ea[wave];
  const float4* pl = (const float4*)(xl + (size_t)s * HID) + lane * 2;
  const float4* pr = (const float4*)(xr + (size_t)d * HID) + lane * 2;
  const float4* pe = (const float4*)(efeat + (size_t)r * HID) + lane * 2;
  const float4* pa = (const float4*)att + lane * 2;   // att flat index == channel index
  float partial = 0.f;
#pragma unroll
  for (int t = 0; t < 2; ++t) {
    float4 va = pl[t], vb = pr[t], ve = pe[t], vw = pa[t];
    float m;
    m = va.x + vb.x + ve.x; m = (m >= 0.f) ? m : NEG_SLOPE * m; partial += m * vw.x;
    m = va.y + vb.y + ve.y; m = (m >= 0.f) ? m : NEG_SLOPE * m; partial += m * vw.y;
    m = va.z + vb.z + ve.z; m = (m >= 0.f) ? m : NEG_SLOPE * m; partial += m * vw.z;
    m = va.w + vb.w + ve.w; m = (m >= 0.f) ? m : NEG_SLOPE * m; partial += m * vw.w;
  }
  // reduce the 8 lanes belonging to each head (64 channels / 8 per lane)
  partial += __shfl_xor(partial, 1, 32);
  partial += __shfl_xor(partial, 2, 32);
  partial += __shfl_xor(partial, 4, 32);
  if ((lane & 7) == 0) {
    int h = lane >> 3;
    score[(size_t)wave * HEADS + h] = partial;
    atomicMax(&segmax[(size_t)d * HEADS + h], fmap(partial));
  }
}

// score[e,h] <- exp(score - segmax[dst,h]);  segsum[dst,h] += exp(...)
__global__ void expsum_kernel(const int* __restrict__ ei, float* __restrict__ score,
                              const unsigned* __restrict__ segmax, float* __restrict__ segsum,
                              int E_) {
  long long idx = (long long)blockIdx.x * blockDim.x + threadIdx.x;
  if (idx >= (long long)E_ * HEADS) return;
  int e = (int)(idx >> 2);
  int h = (int)(idx & 3);
  int d = ei[(size_t)E_ + e];
  float mx = funmap(segmax[(size_t)d * HEADS + h]);
  float ex = expf(score[idx] - mx);
  score[idx] = ex;
  atomicAdd(&segsum[(size_t)d * HEADS + h], ex);
}

// Wave per edge: hout[dst] += alpha[e,h] * xl[src]
__global__ __launch_bounds__(256) void message_kernel(const int* __restrict__ ei,
                                                      const float* __restrict__ score,
                                                      const float* __restrict__ segsum,
                                                      const float* __restrict__ xl,
                                                      float* __restrict__ hout, int E_) {
  int wave = blockIdx.x * 8 + (threadIdx.x >> 5);
  int lane = threadIdx.x & 31;
  if (wave >= E_) return;
  int s = ei[wave];
  int d = ei[(size_t)E_ + wave];
  int h = lane >> 3;
  float al = score[(size_t)wave * HEADS + h] / (segsum[(size_t)d * HEADS + h] + 1e-16f);
  const float4* pl = (const float4*)(xl + (size_t)s * HID) + lane * 2;
  float* po = hout + (size_t)d * HID + lane * 8;
#pragma unroll
  for (int t = 0; t < 2; ++t) {
    float4 va = pl[t];
    atomicAdd(po + t * 4 + 0, al * va.x);
    atomicAdd(po + t * 4 + 1, al * va.y);
    atomicAdd(po + t * 4 + 2, al * va.z);
    atomicAdd(po + t * 4 + 3, al * va.w);
  }
}

__global__ void finalize_kernel(const float* __restrict__ hout, const float* __restrict__ bias,
                                float* __restrict__ h1, long long n) {
  long long idx = (long long)blockIdx.x * blockDim.x + threadIdx.x;
  if (idx >= n) return;
  int f = (int)(idx & (HID - 1));
  float v = hout[idx] + bias[f];
  h1[idx] = v > 0.f ? v : 0.f;
}

__global__ void pool_kernel(const float* __restrict__ h1, const int* __restrict__ batch,
                            float* __restrict__ pool, float* __restrict__ cnt, int N_) {
  long long idx = (long long)blockIdx.x * blockDim.x + threadIdx.x;
  if (idx >= (long long)N_ * HID) return;
  int i = (int)(idx >> 8);
  int f = (int)(idx & (HID - 1));
  int b = batch[i];
  atomicAdd(&pool[(size_t)b * HID + f], h1[idx]);
  if (f == 0) atomicAdd(&cnt[b], 1.0f);
}

__global__ __launch_bounds__(256) void mlp1_kernel(const float* __restrict__ pool,
                                                   const float* __restrict__ cnt,
                                                   const float* __restrict__ W1,
                                                   const float* __restrict__ b1,
                                                   float* __restrict__ y1) {
  int g = blockIdx.x, f = threadIdx.x;
  __shared__ float gs[HID];
  float c = cnt[g];
  if (c < 1.f) c = 1.f;
  gs[f] = pool[(size_t)g * HID + f] / c;
  __syncthreads();
  float acc = b1[f];
#pragma unroll 8
  for (int k = 0; k < HID; ++k) acc += gs[k] * W1[(size_t)k * HID + f];
  y1[(size_t)g * HID + f] = acc > 0.f ? acc : 0.f;
}

__global__ void mlp2_kernel(const float* __restrict__ y1, const float* __restrict__ W2,
                            const float* __restrict__ b2, float* __restrict__ out) {
  int t = blockIdx.x * blockDim.x + threadIdx.x;
  if (t >= GCNT * NCLS) return;
  int g = t >> 3, c = t & 7;
  float acc = b2[c];
#pragma unroll 8
  for (int k = 0; k < HID; ++k) acc += y1[(size_t)g * HID + k] * W2[(size_t)k * NCLS + c];
  out[t] = acc;
}

extern "C" void kernel_launch(void* const* d_in, const int* in_sizes, int n_in,
                              void* d_out, int out_size, void* d_ws, size_t ws_size,
                              hipStream_t stream) {
  (void)out_size; (void)ws_size;
  const float* x = (const float*)d_in[0];
  const int* ei = (const int*)d_in[1];
  const int* ea = (const int*)d_in[2];
  const int* batch = (const int*)d_in[3];
  const int N_ = in_sizes[3];
  const int E_ = in_sizes[1] / 2;

  auto P = [&](int i) { return (const float*)d_in[i]; };
  const float *tok_emb, *bpW, *bpb, *rel_emb, *m1W, *m1b, *m2W, *m2b;
  const float *lWl[3], *lbl[3], *lWr[3], *lbr[3], *lWe[3], *attp[3], *biasp[3];
  if (n_in >= 33 && in_sizes[4] == 128) {
    // JAX alphabetical pytree order: bbox_proj, convs[att,bias,lin_e,lin_l,lin_r], mlp1, mlp2, rel_emb, tok_emb
    bpW = P(4); bpb = P(5);
    for (int li = 0; li < 3; ++li) {
      int b = 6 + li * 7;
      attp[li] = P(b); biasp[li] = P(b + 1); lWe[li] = P(b + 2);
      lWl[li] = P(b + 3); lbl[li] = P(b + 4); lWr[li] = P(b + 5); lbr[li] = P(b + 6);
    }
    m1W = P(27); m1b = P(28); m2W = P(29); m2b = P(30); rel_emb = P(31); tok_emb = P(32);
  } else {
    // dict insertion order
    tok_emb = P(4); bpW = P(5); bpb = P(6); rel_emb = P(7);
    for (int li = 0; li < 3; ++li) {
      int b = 8 + li * 7;
      lWl[li] = P(b); lbl[li] = P(b + 1); lWr[li] = P(b + 2); lbr[li] = P(b + 3);
      lWe[li] = P(b + 4); attp[li] = P(b + 5); biasp[li] = P(b + 6);
    }
    m1W = P(29); m1b = P(30); m2W = P(31); m2b = P(32);
  }

  // Workspace carve-up (fp32 elements; all slice sizes multiples of 4 -> float4-aligned)
  float* ws = (float*)d_ws;
  size_t o = 0;
  float* h0 = ws + o;      o += (size_t)N_ * IN0;
  float* xl = ws + o;      o += (size_t)N_ * HID;
  float* xr = ws + o;      o += (size_t)N_ * HID;
  float* hout = ws + o;    o += (size_t)N_ * HID;
  float* h1 = ws + o;      o += (size_t)N_ * HID;
  float* score = ws + o;   o += (size_t)E_ * HEADS;
  unsigned* segmax = (unsigned*)(ws + o); o += (size_t)N_ * HEADS;
  float* segsum = ws + o;  o += (size_t)N_ * HEADS;
  float* efeat = ws + o;   o += (size_t)NREL * HID;
  float* pool = ws + o;    o += (size_t)GCNT * HID;
  float* cnt = ws + o;     o += GCNT;           // contiguous after pool: zeroed together
  float* y1 = ws + o;      o += (size_t)GCNT * HID;

  const int T = 256;
  const long long nEmb = (long long)N_ * IN0;
  const long long nNH = (long long)N_ * HID;
  const long long nEH = (long long)E_ * HEADS;
  const unsigned gemmBlocks = (unsigned)(((N_ / 16) * 4 + 7) / 8);
  const unsigned edgeBlocks = (unsigned)((E_ + 7) / 8);

  embed_kernel<<<(unsigned)((nEmb + T - 1) / T), T, 0, stream>>>(x, tok_emb, bpW, bpb, h0, N_);

  for (int li = 0; li < 3; ++li) {
    const float* hin = (li == 0) ? h0 : h1;
    const int K = (li == 0) ? IN0 : HID;
    efeat_kernel<<<NREL, 256, 0, stream>>>(rel_emb, lWe[li], efeat);
    gemm_wmma_kernel<<<gemmBlocks, 256, 0, stream>>>(hin, lWl[li], lbl[li], xl, N_, K);
    gemm_wmma_kernel<<<gemmBlocks, 256, 0, stream>>>(hin, lWr[li], lbr[li], xr, N_, K);
    fill_u32_kernel<<<(unsigned)((nEH / HEADS * HEADS / E_ * 0 + (long long)N_ * HEADS + T - 1) / T), T, 0, stream>>>(segmax, (long long)N_ * HEADS, 0u);
    fill_u32_kernel<<<(unsigned)(((long long)N_ * HEADS + T - 1) / T), T, 0, stream>>>((unsigned*)segsum, (long long)N_ * HEADS, 0u);
    fill_u32_kernel<<<(unsigned)((nNH + T - 1) / T), T, 0, stream>>>((unsigned*)hout, nNH, 0u);
    edge_score_kernel<<<edgeBlocks, 256, 0, stream>>>(ei, ea, xl, xr, efeat, attp[li], score, segmax, E_);
    expsum_kernel<<<(unsigned)((nEH + T - 1) / T), T, 0, stream>>>(ei, score, segmax, segsum, E_);
    message_kernel<<<edgeBlocks, 256, 0, stream>>>(ei, score, segsum, xl, hout, E_);
    finalize_kernel<<<(unsigned)((nNH + T - 1) / T), T, 0, stream>>>(hout, biasp[li], h1, nNH);
  }

  fill_u32_kernel<<<(unsigned)((GCNT * HID + GCNT + T - 1) / T), T, 0, stream>>>((unsigned*)pool, GCNT * HID + GCNT, 0u);
  pool_kernel<<<(unsigned)((nNH + T - 1) / T), T, 0, stream>>>(h1, batch, pool, cnt, N_);
  mlp1_kernel<<<GCNT, 256, 0, stream>>>(pool, cnt, m1W, m1b, y1);
  mlp2_kernel<<<2, 256, 0, stream>>>(y1, m2W, m2b, (float*)d_out);
}